// NetworksPlusCircuit_15255723835471
// MI455X (gfx1250) — compile-verified
//
#include <hip/hip_runtime.h>

typedef __attribute__((ext_vector_type(2))) float v2f;
typedef __attribute__((ext_vector_type(8))) float v8f;

#define THREADS_PER_BLOCK 256
#define WAVES_PER_BLOCK   8
#define ROWS_PER_WAVE     32
#define ROWS_PER_BLOCK    256   // WAVES_PER_BLOCK * ROWS_PER_WAVE

// ---------------------------------------------------------------------------
// Fused: (Bx64)@(64x16) fp32 WMMA GEMM -> sigmoid -> 16-literal SDD scan.
// One wave handles 32 rows as two 16x16xK accumulations with
// v_wmma_f32_16x16x4_f32 over K=64 in 16 chunks of 4.
// ---------------------------------------------------------------------------
__global__ __launch_bounds__(THREADS_PER_BLOCK)
void sdd_gemm_circuit_kernel(const float* __restrict__ x,
                             const float* __restrict__ Ws,   // (4,64,4)
                             const float* __restrict__ bs,   // (4,4) flat 16
                             float* __restrict__ out,
                             int numTiles) {
    // W_cat[d][c] = Ws[c>>2][d][c&3], stored [64][16]
    __shared__ float s_w[64 * 16];
    // per-wave transpose buffer, stride 17 -> conflict-free column reads
    __shared__ float s_h[WAVES_PER_BLOCK][32][17];

    const int tid  = threadIdx.x;
    const int wave = tid >> 5;
    const int lane = tid & 31;
    const int lo   = lane & 15;   // N (and M base) within 16
    const int hi   = lane >> 4;   // K-half selector per ISA WMMA layout

    // ---- stage the (64x16) concatenated weight matrix into LDS ------------
    for (int idx = tid; idx < 64 * 16; idx += THREADS_PER_BLOCK) {
        int d = idx >> 4;
        int c = idx & 15;
        s_w[idx] = Ws[((c >> 2) << 8) + (d << 2) + (c & 3)];
    }
    __syncthreads();

    // ---- per-lane B fragments for all 16 K-chunks (ISA 16x4 fp32 layout) --
    // lane<16 holds K = 4t+{0,1}; lane>=16 holds K = 4t+{2,3}; N = lo
    v2f bfrag[16];
#pragma unroll
    for (int t = 0; t < 16; ++t) {
        int k0 = t * 4 + hi * 2;
        bfrag[t].x = s_w[k0 * 16 + lo];
        bfrag[t].y = s_w[(k0 + 1) * 16 + lo];
    }
    const float bias = bs[lo];   // column c bias == bs flat[c]

    const size_t tileStrideFloats = (size_t)gridDim.x * ROWS_PER_BLOCK * 64;

    for (int tile = blockIdx.x; tile < numTiles; tile += gridDim.x) {
        const long rowBase = (long)tile * ROWS_PER_BLOCK + wave * ROWS_PER_WAVE;

        // A fragment base: row = rowBase + (lane&15), col = hi*2 (+ 4t)
        const float* xr0 = x + (size_t)(rowBase + lo) * 64 + hi * 2;
        const float* xr1 = xr0 + 16 * 64;

        // speculative prefetch of next tile's portion (global_prefetch_b8)
        __builtin_prefetch(xr0 + tileStrideFloats, 0, 1);

        v8f c0 = {};
        v8f c1 = {};
#pragma unroll
        for (int t = 0; t < 16; ++t) {
            v2f a0 = *(const v2f*)(xr0 + t * 4);
            v2f a1 = *(const v2f*)(xr1 + t * 4);
            c0 = __builtin_amdgcn_wmma_f32_16x16x4_f32(
                     false, a0, false, bfrag[t], (short)0, c0, false, false);
            c1 = __builtin_amdgcn_wmma_f32_16x16x4_f32(
                     false, a1, false, bfrag[t], (short)0, c1, false, false);
        }

        // ---- sigmoid + transpose through LDS ------------------------------
        // C layout: vgpr r, lanes 0-15: (M=r, N=lane); lanes 16-31: (M=r+8)
#pragma unroll
        for (int r = 0; r < 8; ++r) {
            int m = r + hi * 8;
            float h0 = 1.0f / (1.0f + __expf(-(c0[r] + bias)));
            float h1 = 1.0f / (1.0f + __expf(-(c1[r] + bias)));
            s_h[wave][m][lo]      = h0;
            s_h[wave][16 + m][lo] = h1;
        }
        __syncthreads();

        // ---- per-row SDD circuit scan (each lane = one of 32 rows) --------
        float v[16];
#pragma unroll
        for (int c = 0; c < 16; ++c) v[c] = s_h[wave][lane][c];

        float fp1 = 1.0f, fp2 = 1.0f;   // f(i+1), f(i+2)
#pragma unroll
        for (int i = 16; i >= 1; --i) {
            float pos = v[i - 1];
            float neg = (i == 3 || i == 7) ? 1.0f : (1.0f - pos);
            float fi  = fmaf(pos, fp1, neg * fp2);
            fp2 = fp1;
            fp1 = fi;
        }
        out[rowBase + lane] = fp1;
        __syncthreads();   // protect s_h before next iteration's writes
    }
}

// ---------------------------------------------------------------------------
// Scalar tail for rows not covered by full 256-row tiles (B % 256 != 0).
// Unused for B = 1M but keeps the launch total & deterministic.
// ---------------------------------------------------------------------------
__global__ void sdd_tail_kernel(const float* __restrict__ x,
                                const float* __restrict__ Ws,
                                const float* __restrict__ bs,
                                float* __restrict__ out,
                                long rowStart, long B) {
    long row = rowStart + blockIdx.x * (long)blockDim.x + threadIdx.x;
    if (row >= B) return;

    const float* xr = x + row * 64;
    float v[16];
#pragma unroll
    for (int c = 0; c < 16; ++c) {
        int n = c >> 2, k = c & 3;
        float acc = 0.0f;
        for (int d = 0; d < 64; ++d)
            acc = fmaf(xr[d], Ws[(n << 8) + (d << 2) + k], acc);
        v[c] = 1.0f / (1.0f + __expf(-(acc + bs[c])));
    }
    float fp1 = 1.0f, fp2 = 1.0f;
#pragma unroll
    for (int i = 16; i >= 1; --i) {
        float pos = v[i - 1];
        float neg = (i == 3 || i == 7) ? 1.0f : (1.0f - pos);
        float fi  = fmaf(pos, fp1, neg * fp2);
        fp2 = fp1;
        fp1 = fi;
    }
    out[row] = fp1;
}

extern "C" void kernel_launch(void* const* d_in, const int* in_sizes, int n_in,
                              void* d_out, int out_size, void* d_ws, size_t ws_size,
                              hipStream_t stream) {
    const float* x  = (const float*)d_in[0];   // (B, 64) f32
    const float* Ws = (const float*)d_in[1];   // (4, 64, 4) f32
    const float* bs = (const float*)d_in[2];   // (4, 4) f32
    float* out = (float*)d_out;                // (B,) f32

    const long B = (long)in_sizes[0] / 64;
    const int numTiles = (int)(B / ROWS_PER_BLOCK);

    if (numTiles > 0) {
        int grid = numTiles < 2048 ? numTiles : 2048;
        sdd_gemm_circuit_kernel<<<grid, THREADS_PER_BLOCK, 0, stream>>>(
            x, Ws, bs, out, numTiles);
    }
    const long rem = B - (long)numTiles * ROWS_PER_BLOCK;
    if (rem > 0) {
        int grid = (int)((rem + 255) / 256);
        sdd_tail_kernel<<<grid, 256, 0, stream>>>(x, Ws, bs, out,
                                                  (long)numTiles * ROWS_PER_BLOCK, B);
    }
}